// DiffGCN_67963562492651
// MI455X (gfx1250) — compile-verified
//
#include <hip/hip_runtime.h>
#include <hip/hip_bf16.h>

#define N_NODES 8000
#define C_IN    64
#define DEG     16
#define HID     128
#define C_OUTD  64
#define T_STEPS 4
#define G3      384   // 3*HID

typedef __attribute__((ext_vector_type(2))) float v2f;
typedef __attribute__((ext_vector_type(8))) float v8f;

__device__ __forceinline__ float sigmoidf_(float x) { return 1.0f / (1.0f + expf(-x)); }

// C[M,Ncols] = A[M,K] @ B[K,Ncols] + bias[Ncols], fp32 WMMA 16x16x4.
// blockDim.x = 128 (4 waves). Each wave computes one 16x16 tile.
// grid: (M/16, Ncols/64). Grid sized exactly -> EXEC all-1s for WMMA.
__global__ void wmma_gemm_bias(const float* __restrict__ A, const float* __restrict__ B,
                               const float* __restrict__ bias, float* __restrict__ Cm,
                               int K, int Ncols) {
    const int lane = threadIdx.x & 31;
    const int wave = threadIdx.x >> 5;
    const int row0 = blockIdx.x << 4;
    const int col0 = ((blockIdx.y << 2) + wave) << 4;
    const int m    = lane & 15;
    const int half = lane >> 4;

    const float* arow = A + (size_t)(row0 + m) * K;
    v8f acc = {};
    for (int k = 0; k < K; k += 4) {
        const int ka = k + (half << 1);          // K = 0/1 (lanes 0-15), 2/3 (lanes 16-31)
        v2f a, b;
        a.x = arow[ka];
        a.y = arow[ka + 1];
        b.x = B[(size_t)ka * Ncols + col0 + m];
        b.y = B[(size_t)(ka + 1) * Ncols + col0 + m];
        acc = __builtin_amdgcn_wmma_f32_16x16x4_f32(false, a, false, b, (short)0, acc,
                                                    false, false);
    }
    const float bv = bias[col0 + m];
#pragma unroll
    for (int v = 0; v < 8; ++v) {                // D: VGPR v -> row v (lanes 0-15) / v+8
        Cm[(size_t)(row0 + v + (half << 3)) * Ncols + col0 + m] = acc[v] + bv;
    }
}

// One GRU step, elementwise over (node, h). If (flags&1): h_old=0, hg=bh (first token).
// ids!=nullptr: gather xg row via ids[i*ids_stride]. (flags&2): init cur/walk_ids.
__global__ void gru_elem(const float* __restrict__ XG, const int* __restrict__ ids,
                         int ids_stride, const float* __restrict__ HG,
                         const float* __restrict__ bh, float* __restrict__ h,
                         int* __restrict__ cur, int* __restrict__ walk_ids, int flags) {
    int idx = blockIdx.x * blockDim.x + threadIdx.x;
    if (idx >= N_NODES * HID) return;
    const int i  = idx >> 7;
    const int hh = idx & (HID - 1);
    const int row = ids ? ids[(size_t)i * ids_stride] : i;
    const float* xrow = XG + (size_t)row * G3;
    const float xr = xrow[hh], xz = xrow[HID + hh], xn = xrow[2 * HID + hh];
    float hr, hz, hn, hold;
    if (flags & 1) {
        hr = bh[hh]; hz = bh[HID + hh]; hn = bh[2 * HID + hh]; hold = 0.0f;
    } else {
        const float* grow = HG + (size_t)i * G3;
        hr = grow[hh]; hz = grow[HID + hh]; hn = grow[2 * HID + hh];
        hold = h[idx];
    }
    const float r = sigmoidf_(xr + hr);
    const float z = sigmoidf_(xz + hz);
    const float n = tanhf(xn + r * hn);
    h[idx] = (1.0f - z) * n + z * hold;
    if ((flags & 2) && hh == 0) { cur[i] = i; walk_ids[i * 5] = i; }
}

// One wave (32 lanes) per node: score 16 candidates (shared hg + gathered xg),
// softmax-normalize, perturb with noise, argmax (first-max, matching jnp.argmax),
// emit walks_logp[:,t], advance cur / walk_ids / h_d.
__global__ void select_step(const float* __restrict__ XG, const float* __restrict__ HG,
                            float* __restrict__ hd, const int* __restrict__ neighbors,
                            int* __restrict__ cur, int* __restrict__ walk_ids,
                            const float* __restrict__ noise_t, const float* __restrict__ dWo,
                            const float* __restrict__ dbo, float* __restrict__ logp_out,
                            int t) {
    const int lane = threadIdx.x & 31;
    const int wave = threadIdx.x >> 5;
    const int i = blockIdx.x * 8 + wave;        // grid exact: 1000*8 = 8000
    const int ci = cur[i];

    float hr[4], hz[4], hn[4], hold[4], wo[4];
    const float* grow = HG + (size_t)i * G3;
    const float* hrow = hd + (size_t)i * HID;
#pragma unroll
    for (int q = 0; q < 4; ++q) {
        const int hh = (q << 5) + lane;
        hr[q]  = grow[hh];
        hz[q]  = grow[HID + hh];
        hn[q]  = grow[2 * HID + hh];
        hold[q] = hrow[hh];
        wo[q]  = dWo[(size_t)hh * C_OUTD];      // column 0 of dWo
    }
    const float bo0 = dbo[0];

    float logp[DEG];
#pragma unroll
    for (int j = 0; j < DEG; ++j) {
        const int nb = neighbors[(size_t)ci * DEG + j];
        const float* xrow = XG + (size_t)nb * G3;
        float dot = 0.0f;
#pragma unroll
        for (int q = 0; q < 4; ++q) {
            const int hh = (q << 5) + lane;
            const float r = sigmoidf_(xrow[hh] + hr[q]);
            const float z = sigmoidf_(xrow[HID + hh] + hz[q]);
            const float n = tanhf(xrow[2 * HID + hh] + r * hn[q]);
            const float hnew = (1.0f - z) * n + z * hold[q];
            dot = fmaf(hnew, wo[q], dot);
        }
#pragma unroll
        for (int off = 16; off >= 1; off >>= 1) dot += __shfl_xor(dot, off, 32);
        logp[j] = dot + bo0;                    // broadcast to all lanes
    }

    float mx = logp[0];
#pragma unroll
    for (int j = 1; j < DEG; ++j) mx = fmaxf(mx, logp[j]);
    float s = 0.0f;
#pragma unroll
    for (int j = 0; j < DEG; ++j) s += expf(logp[j] - mx);
    const float norm = mx + logf(s);

    int sel = 0; float best = -3.4e38f;
#pragma unroll
    for (int j = 0; j < DEG; ++j) {
        const float noisy = expf(logp[j] - norm) + 0.01f * noise_t[(size_t)i * DEG + j];
        if (noisy > best) { best = noisy; sel = j; }   // strict > : first max wins
    }
    const int newcur = neighbors[(size_t)ci * DEG + sel];
    if (lane == 0) {
        cur[i] = newcur;
        walk_ids[i * 5 + (t + 1)] = newcur;
        logp_out[(size_t)i * T_STEPS + t] = logp[sel] - norm;
    }
    // Selected candidate's h_new == next prefix hidden state.
    const float* xrow = XG + (size_t)newcur * G3;
#pragma unroll
    for (int q = 0; q < 4; ++q) {
        const int hh = (q << 5) + lane;
        const float r = sigmoidf_(xrow[hh] + hr[q]);
        const float z = sigmoidf_(xrow[HID + hh] + hz[q]);
        const float n = tanhf(xrow[2 * HID + hh] + r * hn[q]);
        hd[(size_t)i * HID + hh] = (1.0f - z) * n + z * hold[q];
    }
}

extern "C" void kernel_launch(void* const* d_in, const int* in_sizes, int n_in,
                              void* d_out, int out_size, void* d_ws, size_t ws_size,
                              hipStream_t stream) {
    const float* node_attr = (const float*)d_in[0];
    const int*   neighbors = (const int*)d_in[1];
    const float* noise     = (const float*)d_in[2];
    const float* dWx = (const float*)d_in[3];
    const float* dWh = (const float*)d_in[4];
    const float* dbx = (const float*)d_in[5];
    const float* dbh = (const float*)d_in[6];
    const float* dWo = (const float*)d_in[7];
    const float* dbo = (const float*)d_in[8];
    const float* wWx = (const float*)d_in[9];
    const float* wWh = (const float*)d_in[10];
    const float* wbx = (const float*)d_in[11];
    const float* wbh = (const float*)d_in[12];
    const float* wWo = (const float*)d_in[13];
    const float* wbo = (const float*)d_in[14];
    (void)in_sizes; (void)n_in; (void)out_size; (void)ws_size;

    char* ws = (char*)d_ws;
    size_t off = 0;
    auto carve = [&](size_t bytes) {
        void* p = ws + off;
        off = (off + bytes + 255) & ~(size_t)255;
        return p;
    };
    float* XGd = (float*)carve((size_t)N_NODES * G3 * sizeof(float));  // node_attr@dWx+dbx
    float* XGw = (float*)carve((size_t)N_NODES * G3 * sizeof(float));  // node_attr@wWx+wbx
    float* HG  = (float*)carve((size_t)N_NODES * G3 * sizeof(float));  // h @ Wh + bh (reused)
    float* hd  = (float*)carve((size_t)N_NODES * HID * sizeof(float));
    float* hw  = (float*)carve((size_t)N_NODES * HID * sizeof(float));
    int* cur      = (int*)carve((size_t)N_NODES * sizeof(int));
    int* walk_ids = (int*)carve((size_t)N_NODES * 5 * sizeof(int));

    float* v_out    = (float*)d_out;                       // [N,64]
    float* logp_out = v_out + (size_t)N_NODES * C_OUTD;    // [N,4]

    const dim3 gblk(128);
    const int ethreads = N_NODES * HID;
    const int eblocks  = (ethreads + 255) / 256;

    // xg precompute (both GRUs): [8000,64] @ [64,384]
    wmma_gemm_bias<<<dim3(N_NODES / 16, G3 / 64), gblk, 0, stream>>>(node_attr, dWx, dbx, XGd, C_IN, G3);
    wmma_gemm_bias<<<dim3(N_NODES / 16, G3 / 64), gblk, 0, stream>>>(node_attr, wWx, wbx, XGw, C_IN, G3);

    // prefix hidden after token0 (h0 = 0 => hg = dbh); init cur / walk_ids
    gru_elem<<<eblocks, 256, 0, stream>>>(XGd, nullptr, 0, nullptr, dbh, hd, cur, walk_ids, 3);

    for (int t = 0; t < T_STEPS; ++t) {
        // shared hg for all 16 candidates: [8000,128] @ [128,384]
        wmma_gemm_bias<<<dim3(N_NODES / 16, G3 / 64), gblk, 0, stream>>>(hd, dWh, dbh, HG, HID, G3);
        select_step<<<N_NODES / 8, 256, 0, stream>>>(XGd, HG, hd, neighbors, cur, walk_ids,
                                                     noise + (size_t)t * N_NODES * DEG,
                                                     dWo, dbo, logp_out, t);
    }

    // final GRU over the 5 walk tokens (token0 = node itself)
    gru_elem<<<eblocks, 256, 0, stream>>>(XGw, nullptr, 0, nullptr, wbh, hw, cur, walk_ids, 1);
    for (int s = 1; s <= T_STEPS; ++s) {
        wmma_gemm_bias<<<dim3(N_NODES / 16, G3 / 64), gblk, 0, stream>>>(hw, wWh, wbh, HG, HID, G3);
        gru_elem<<<eblocks, 256, 0, stream>>>(XGw, walk_ids + s, 5, HG, wbh, hw, cur, walk_ids, 0);
    }

    // v_out = h_w @ wWo + wbo : [8000,128] @ [128,64]
    wmma_gemm_bias<<<dim3(N_NODES / 16, C_OUTD / 64), gblk, 0, stream>>>(hw, wWo, wbo, v_out, HID, C_OUTD);
}